// MultiHeadAttention_44306882625979
// MI455X (gfx1250) — compile-verified
//
#include <hip/hip_runtime.h>

typedef __attribute__((ext_vector_type(2))) float v2f;
typedef __attribute__((ext_vector_type(8))) float v8f;

#define Bb   2
#define Cc   512
#define HWsz 4096   // 64*64

// ---------------------------------------------------------------------------
// K1: per-(b,c) InstanceNorm statistics over 4096 spatial elements.
// ---------------------------------------------------------------------------
__global__ void k_instnorm_stats(const float* __restrict__ x,
                                 float* __restrict__ mu_out,
                                 float* __restrict__ rs_out) {
    const int bid = blockIdx.x;                      // 0..1023 = b*512 + c
    const float4* xp = (const float4*)(x + (size_t)bid * HWsz);
    float s = 0.f, s2 = 0.f;
    for (int i = threadIdx.x; i < HWsz / 4; i += 256) {
        float4 v = xp[i];
        s  += v.x + v.y + v.z + v.w;
        s2 += v.x * v.x + v.y * v.y + v.z * v.z + v.w * v.w;
    }
    for (int m = 16; m > 0; m >>= 1) {
        s  += __shfl_xor(s,  m, 32);
        s2 += __shfl_xor(s2, m, 32);
    }
    __shared__ float sa[8], sb[8];
    const int w = threadIdx.x >> 5, l = threadIdx.x & 31;
    if (l == 0) { sa[w] = s; sb[w] = s2; }
    __syncthreads();
    if (threadIdx.x == 0) {
        float S = 0.f, S2 = 0.f;
        for (int i = 0; i < 8; ++i) { S += sa[i]; S2 += sb[i]; }
        float m  = S  * (1.f / HWsz);
        float var = S2 * (1.f / HWsz) - m * m;
        mu_out[bid] = m;
        rs_out[bid] = rsqrtf(var + 1e-5f);
    }
}

// ---------------------------------------------------------------------------
// K1b: zero the colsum accumulator (ws is poisoned by the harness).
// ---------------------------------------------------------------------------
__global__ void k_zero(float* __restrict__ p, int n) {
    int i = blockIdx.x * blockDim.x + threadIdx.x;
    if (i < n) p[i] = 0.f;
}

// ---------------------------------------------------------------------------
// K2: V-projection GEMM via V_WMMA_F32_16X16X4_F32, fused with:
//   - on-the-fly InstanceNorm normalization of x (stats staged in LDS)
//   - spatial column-sum reduction of v (shuffle -> LDS -> global atomic)
// v[o,hw] = sum_c w_qkv[1024+o, c] * (x[b,c,hw]-mu)*rsig ;  colsum[b,o] += v
// Block: 256 threads = 8 waves. blockIdx = (hw_group 0..31, o_tile 0..31, b).
// Each wave: one 16(o) x 16(hw) tile, K-loop over 512 channels (128 WMMA).
// ---------------------------------------------------------------------------
__global__ void __launch_bounds__(256)
k_vcolsum_wmma(const float* __restrict__ x,
               const float* __restrict__ w_qkv,
               const float* __restrict__ mu,
               const float* __restrict__ rs,
               float* __restrict__ colsum) {
    __shared__ float smu[Cc];
    __shared__ float srs[Cc];
    __shared__ float lds_col[16];

    const int tid  = threadIdx.x;
    const int b    = blockIdx.z;
    const int o0   = blockIdx.y * 16;                 // v-channel tile base
    const int wid  = tid >> 5;
    const int lane = tid & 31;

    for (int i = tid; i < Cc; i += 256) {
        smu[i] = mu[b * Cc + i];
        srs[i] = rs[b * Cc + i];
    }
    if (tid < 16) lds_col[tid] = 0.f;
    __syncthreads();

    const int m     = lane & 15;                      // A row within tile / B col
    const int halfk = (lane >> 4) * 2;                // K sub-offset per ISA layout
    const int hw    = blockIdx.x * 128 + wid * 16 + m;

    // A: rows of w_qkv for the V part (channel offset 2C = 1024)
    const float* __restrict__ Arow = w_qkv + (size_t)(1024 + o0 + m) * Cc;
    // B: x at fixed spatial position, variable channel (stride HWsz)
    const float* __restrict__ xp   = x + ((size_t)b << 21) + hw;

    v8f acc = {};
#pragma unroll 4
    for (int k0 = 0; k0 < Cc; k0 += 4) {
        const int ca = k0 + halfk;
        float2 af = *(const float2*)(Arow + ca);
        float b0 = (xp[(size_t)ca * HWsz]       - smu[ca])     * srs[ca];
        float b1 = (xp[(size_t)(ca + 1) * HWsz] - smu[ca + 1]) * srs[ca + 1];
        v2f av = {af.x, af.y};
        v2f bv = {b0, b1};
        acc = __builtin_amdgcn_wmma_f32_16x16x4_f32(
            /*neg_a=*/false, av, /*neg_b=*/false, bv,
            /*c_mod=*/(short)0, acc, /*reuse_a=*/false, /*reuse_b=*/false);
    }

    // Reduce tile over the 16 hw columns: D layout = VGPR r, lanes0-15:M=r,
    // lanes16-31:M=r+8. XOR-shuffles 1,2,4,8 reduce within each 16-lane half.
#pragma unroll
    for (int r = 0; r < 8; ++r) {
        float v = acc[r];
        v += __shfl_xor(v, 1, 32);
        v += __shfl_xor(v, 2, 32);
        v += __shfl_xor(v, 4, 32);
        v += __shfl_xor(v, 8, 32);
        if ((lane & 15) == 0)                           // lanes 0 and 16
            atomicAdd(&lds_col[r + ((lane >> 4) << 3)], v);
    }
    __syncthreads();
    if (tid < 16) atomicAdd(&colsum[b * Cc + o0 + tid], lds_col[tid]);
}

// ---------------------------------------------------------------------------
// K3: wsum[o] = sum_c w_proj[o,c]  and  G[b,e] = colsum_v + 4096*bias,
//     with e = d*8+n  =>  v-channel o = (e%8)*64 + e/8.
// ---------------------------------------------------------------------------
__global__ void k_wsum_G(const float* __restrict__ w_proj,
                         const float* __restrict__ b_qkv,
                         const float* __restrict__ colsum,
                         float* __restrict__ wsum,
                         float* __restrict__ G) {
    int gid = blockIdx.x * blockDim.x + threadIdx.x;
    if (gid < Cc) {
        const float* row = w_proj + (size_t)gid * Cc;
        float s = 0.f;
        for (int c = 0; c < Cc; ++c) s += row[c];
        wsum[gid] = s;
    } else if (gid < Cc + Bb * Cc) {
        int t = gid - Cc;
        int b = t >> 9, e = t & 511;
        int o = ((e & 7) << 6) + (e >> 3);            // n*64 + d
        G[b * Cc + e] = colsum[b * Cc + o] + (float)HWsz * b_qkv[1024 + o];
    }
}

// ---------------------------------------------------------------------------
// K4: fused epilogue: out[b,o,s] = x + b_proj[o] + wsum[o] * G[b, s%512]
// (the 1x1 output conv is rank-1 because g is c-independent). float4, HBM-bound.
// ---------------------------------------------------------------------------
__global__ void k_final(const float* __restrict__ x,
                        const float* __restrict__ b_proj,
                        const float* __restrict__ wsum,
                        const float* __restrict__ G,
                        float* __restrict__ out) {
    const int i   = blockIdx.x * 256 + threadIdx.x;   // float4 index
    const int idx = i << 2;
    const int b   = idx >> 21;
    const int rem = idx & ((1 << 21) - 1);
    const int o   = rem >> 12;
    const int e0  = rem & 511;                         // s % 512 (4-aligned)

    float4 xv = ((const float4*)x)[i];
    float4 gv = *(const float4*)(G + b * Cc + e0);
    float  bp = b_proj[o];
    float  wv = wsum[o];

    float4 r;
    r.x = xv.x + bp + wv * gv.x;
    r.y = xv.y + bp + wv * gv.y;
    r.z = xv.z + bp + wv * gv.z;
    r.w = xv.w + bp + wv * gv.w;
    ((float4*)out)[i] = r;
}

// ---------------------------------------------------------------------------
extern "C" void kernel_launch(void* const* d_in, const int* in_sizes, int n_in,
                              void* d_out, int out_size, void* d_ws, size_t ws_size,
                              hipStream_t stream) {
    const float* x      = (const float*)d_in[0];
    const float* w_qkv  = (const float*)d_in[1];
    const float* b_qkv  = (const float*)d_in[2];
    const float* w_proj = (const float*)d_in[3];
    const float* b_proj = (const float*)d_in[4];
    float* out = (float*)d_out;

    float* ws     = (float*)d_ws;
    float* mu     = ws;              // 1024
    float* rs     = ws + 1024;       // 1024
    float* colsum = ws + 2048;       // 1024
    float* wsum   = ws + 3072;       // 512
    float* G      = ws + 3584;       // 1024

    k_instnorm_stats<<<Bb * Cc, 256, 0, stream>>>(x, mu, rs);
    k_zero<<<4, 256, 0, stream>>>(colsum, Bb * Cc);
    k_vcolsum_wmma<<<dim3(HWsz / 128, Cc / 16, Bb), 256, 0, stream>>>(
        x, w_qkv, mu, rs, colsum);
    k_wsum_G<<<6, 256, 0, stream>>>(w_proj, b_qkv, colsum, wsum, G);
    k_final<<<(Bb * Cc * HWsz) / 4 / 256, 256, 0, stream>>>(x, b_proj, wsum, G, out);
}